// CTCDecoder_30966714204687
// MI455X (gfx1250) — compile-verified
//
#include <hip/hip_runtime.h>
#include <hip/hip_bf16.h>
#include <math.h>

// CTC best-path decode for B=64, T=256, C=2500 (blank = C-1).
// Phase 1: per-frame argmax over C, one wave32 per frame, 4-deep async-to-LDS ring.
// Phase 2: per-batch sequential compaction (merge repeats, drop blank) + score.
//
// Roofline: 64*256*2500*4 B = 163.8 MB single-pass read -> ~7.0 us at 23.3 TB/s.

#define B_   64
#define T_   256
#define C_   2500
#define NQ   (C_ / 4)            // 625 float4 per frame (2500 % 4 == 0)
#define BLANK (C_ - 1)
#define WAVES_PER_BLOCK 8
#define NCHUNK ((NQ + 31) / 32)  // 20 chunks of 32 float4 per wave
#define DEPTH 4                  // async ring depth (chunks in flight per wave)

__global__ __launch_bounds__(256)
void ctc_argmax_kernel(const float* __restrict__ probs,
                       int* __restrict__ bestIdxOut,
                       float* __restrict__ maxLpOut)
{
    // per-wave ring: 4 x 32 lanes x 16B = 2 KB per wave, 16 KB per block
    __shared__ float4 buf[WAVES_PER_BLOCK][DEPTH][32];

    const int lane  = threadIdx.x & 31;
    const int wave  = threadIdx.x >> 5;
    const int frame = blockIdx.x * WAVES_PER_BLOCK + wave;   // grid exactly covers B*T

    const float4* src = (const float4*)(probs + (size_t)frame * C_); // 16B aligned

    // Branch-free issue: lanes past the end re-load the last float4 (q clamped).
    // The clamped lanes compute *correct* indices for that duplicated data, so
    // the tie->lowest-index argmax reduction is unaffected. No EXEC divergence.
    auto issue = [&](int c) {
        int q = c * 32 + lane;
        q = q < NQ ? q : NQ - 1;
        unsigned lds          = (unsigned)(size_t)&buf[wave][c & (DEPTH - 1)][lane];
        unsigned long long ga = (unsigned long long)(size_t)(src + q);
        // s_wait_dscnt 0: the slot being refilled was ds_load'ed this iteration;
        // ensure that read has fully completed before the async write can land.
        asm volatile("s_wait_dscnt 0x0\n\t"
                     "global_load_async_to_lds_b128 %0, %1, off"
                     :: "v"(lds), "v"(ga) : "memory");
    };

    float best    = -1.0f;   // softmax outputs are >= 0, so -1 is below all
    int   bestIdx = 0;

    auto consume = [&](int c) {
        float4 v = buf[wave][c & (DEPTH - 1)][lane];   // ds_load_b128
        int q = c * 32 + lane;
        q = q < NQ ? q : NQ - 1;
        int k0 = q * 4;
        // within the 4 packed values: tie -> lowest class index (strict >)
        float m = v.x; int mi = k0;
        if (v.y > m) { m = v.y; mi = k0 + 1; }
        if (v.z > m) { m = v.z; mi = k0 + 2; }
        if (v.w > m) { m = v.w; mi = k0 + 3; }
        // within a lane, q ascends with c: strict > keeps the earliest index
        if (m > best) { best = m; bestIdx = mi; }
    };

    issue(0); issue(1); issue(2); issue(3);

#pragma unroll 4
    for (int c = 0; c < NCHUNK - DEPTH; ++c) {          // c = 0..15, fully uniform
        asm volatile("s_wait_asynccnt 0x3" ::: "memory"); // chunk c landed (c+1..c+3 in flight)
        consume(c);
        issue(c + DEPTH);
    }
    // drain: outstanding count steps down 4 -> 1
    asm volatile("s_wait_asynccnt 0x3" ::: "memory"); consume(NCHUNK - 4);
    asm volatile("s_wait_asynccnt 0x2" ::: "memory"); consume(NCHUNK - 3);
    asm volatile("s_wait_asynccnt 0x1" ::: "memory"); consume(NCHUNK - 2);
    asm volatile("s_wait_asynccnt 0x0" ::: "memory"); consume(NCHUNK - 1);

    // wave32 argmax reduction, tie -> lowest index (matches jnp.argmax)
    for (int off = 16; off > 0; off >>= 1) {
        float ov = __shfl_xor(best, off, 32);
        int   oi = __shfl_xor(bestIdx, off, 32);
        if (ov > best || (ov == best && oi < bestIdx)) { best = ov; bestIdx = oi; }
    }

    if (lane == 0) {
        bestIdxOut[frame] = bestIdx;
        maxLpOut[frame]   = logf(best + 1e-7f);  // log monotone: argmax(p) == argmax(log(p+eps))
    }
}

__global__ __launch_bounds__(64)
void ctc_compact_kernel(const int* __restrict__ bestIdx,
                        const float* __restrict__ maxLp,
                        float* __restrict__ out)
{
    int b = blockIdx.x * blockDim.x + threadIdx.x;
    if (b >= B_) return;

    const int*   bi  = bestIdx + b * T_;
    const float* ml  = maxLp   + b * T_;
    float*       dec = out + (size_t)b * T_;

    float sum = 0.0f;
    int pos = 0, prev = -1;
#pragma unroll 8
    for (int t = 0; t < T_; ++t) {
        int v = bi[t];
        sum += ml[t];
        if (v != BLANK && v != prev) dec[pos++] = (float)v;
        prev = v;
    }
    for (; pos < T_; ++pos) dec[pos] = -1.0f;

    out[(size_t)B_ * T_ + b] = -sum;   // scores [B,1], flattened after decoded [B,T]
}

extern "C" void kernel_launch(void* const* d_in, const int* in_sizes, int n_in,
                              void* d_out, int out_size, void* d_ws, size_t ws_size,
                              hipStream_t stream)
{
    const float* probs = (const float*)d_in[0];

    // workspace: int best[B*T] followed by float maxlp[B*T]  (128 KB total)
    int*   bestIdx = (int*)d_ws;
    float* maxLp   = (float*)((char*)d_ws + sizeof(int) * (size_t)B_ * T_);

    ctc_argmax_kernel<<<dim3((B_ * T_) / WAVES_PER_BLOCK), dim3(256), 0, stream>>>(
        probs, bestIdx, maxLp);

    ctc_compact_kernel<<<dim3(1), dim3(64), 0, stream>>>(
        bestIdx, maxLp, (float*)d_out);
}